// RowWiseTop_K_Average_PoolingLayer_22041772163402
// MI455X (gfx1250) — compile-verified
//
#include <hip/hip_runtime.h>

#define MIN_VAL (-4294967295.0f)  // -(2^32 - 1)

constexpr int Bn = 32, Xn = 128, Yn = 256, Cn = 64;
constexpr int CHUNK_ROWS  = 16;                  // y rows per LDS chunk (4 KB)
constexpr int CHUNK_BYTES = CHUNK_ROWS * Cn * 4; // 4096
constexpr int NCHUNK      = Yn / CHUNK_ROWS;     // 16
constexpr int NBUF        = 4;                   // 4-deep rotation, 1 barrier/chunk

__device__ __forceinline__ void ins3(float v, float& m1, float& m2, float& m3) {
  if (v > m3) {
    if (v > m2) {
      m3 = m2;
      if (v > m1) { m2 = m1; m1 = v; }
      else        { m2 = v; }
    } else {
      m3 = v;
    }
  }
}

// gfx1250 async global->LDS copy, 16 bytes per lane, GVS addressing:
// mem_addr = SGPR64 base + 32-bit per-lane VGPR offset. VDST = LDS byte addr.
__device__ __forceinline__ void async_copy_b128(uint32_t lds_off, uint32_t voff,
                                                uint64_t sbase) {
  asm volatile("global_load_async_to_lds_b128 %0, %1, %2"
               :: "v"(lds_off), "v"(voff), "s"(sbase)
               : "memory");
}

__global__ __launch_bounds__(256) void topk_pool_kernel(
    const float* __restrict__ inp,     // (B, X, Y, C)
    const float* __restrict__ xmask,   // (B, X)
    const float* __restrict__ ymask,   // (B, Y)
    float* __restrict__ out)           // (B, X, 3, C)
{
  __shared__ __align__(16) float s_buf[NBUF][CHUNK_ROWS * Cn]; // 16 KB rotation
  __shared__ float s_ym[Yn];                                   // 1 KB y_mask
  __shared__ __align__(16) float s_part[4][Cn][4];             // sum,m1,m2,m3
  __shared__ float s_yl[4];                                    // ylen partials

  const int tid = threadIdx.x;
  const int bx  = blockIdx.x;
  const int b   = bx >> 7;        // / X
  const int x   = bx & (Xn - 1);

  // Stage y_mask for this batch (blockDim == Y == 256).
  s_ym[tid] = ymask[b * Yn + tid];

  // Loader role: each 16-lane group copies one contiguous 256B row per chunk.
  const int ysub = tid >> 4;       // 0..15 row-in-chunk
  const int cq   = tid & 15;       // 0..15 float4 column
  // Compute role: thread owns one channel, a quarter of the chunk rows.
  const int ch   = tid & (Cn - 1); // 0..63
  const int q    = tid >> 6;       // 0..3

  const uint64_t sbase =
      (uint64_t)(uintptr_t)(inp + ((size_t)(b * Xn + x) * Yn) * Cn);
  const uint32_t voff = (uint32_t)(ysub * Cn * 4 + cq * 16);

  // Prime the pipeline: chunks 0..2 into buffers 0..2.
  #pragma unroll
  for (int p = 0; p < NBUF - 1; ++p) {
    async_copy_b128((uint32_t)(uintptr_t)&s_buf[p][ysub * Cn + cq * 4],
                    voff + (uint32_t)(p * CHUNK_BYTES), sbase);
  }

  float sum = 0.0f, yl = 0.0f;
  float m1 = MIN_VAL, m2 = MIN_VAL, m3 = MIN_VAL;

  #pragma unroll
  for (int ck = 0; ck < NCHUNK; ++ck) {
    // Chunk ck complete when outstanding <= min(2, 15-ck) (in-order returns).
    if (ck < NCHUNK - 2)       asm volatile("s_wait_asynccnt 0x2" ::: "memory");
    else if (ck == NCHUNK - 2) asm volatile("s_wait_asynccnt 0x1" ::: "memory");
    else                       asm volatile("s_wait_asynccnt 0x0" ::: "memory");
    // One barrier: (a) chunk ck visible from all waves, (b) all waves are done
    // reading buf[(ck-1)&3] == buf[(ck+3)&3], so it may be re-filled now.
    __syncthreads();
    if (ck + NBUF - 1 < NCHUNK) {
      const int nxt = ck + NBUF - 1;
      async_copy_b128(
          (uint32_t)(uintptr_t)&s_buf[nxt & (NBUF - 1)][ysub * Cn + cq * 4],
          voff + (uint32_t)(nxt * CHUNK_BYTES), sbase);
    }

    // Reduce 4 rows of this thread's channel (LDS bank == ch: conflict-free).
    const float* buf = s_buf[ck & (NBUF - 1)];
    #pragma unroll
    for (int r = 0; r < 4; ++r) {
      const int row = q * 4 + r;
      const float ym = s_ym[ck * CHUNK_ROWS + row];
      const float v  = buf[row * Cn + ch];
      sum += v * ym;                                   // mm_zero path
      yl  += ym;                                       // y_real_len partial
      const float cand = (ym != 0.0f) ? v : MIN_VAL;   // mm_min path
      ins3(cand, m1, m2, m3);
    }
  }

  *(float4*)&s_part[q][ch][0] = make_float4(sum, m1, m2, m3);
  if (ch == 0) s_yl[q] = yl;
  __syncthreads();

  if (tid < Cn) {
    float total = 0.0f;
    float M1 = MIN_VAL, M2 = MIN_VAL, M3 = MIN_VAL;
    #pragma unroll
    for (int qq = 0; qq < 4; ++qq) {
      const float4 p = *(const float4*)&s_part[qq][tid][0];
      total += p.x;
      ins3(p.y, M1, M2, M3);
      ins3(p.z, M1, M2, M3);
      ins3(p.w, M1, M2, M3);
    }
    const float ylen = s_yl[0] + s_yl[1] + s_yl[2] + s_yl[3];

    const float xm   = xmask[b * Xn + x];
    const float mean = (ylen > 0.0f) ? (total / ylen) : 0.0f;
    const float o0 = ((ylen < 1.0f) ? mean : M1) * xm;                       // k=1
    const float o1 = ((ylen < 3.0f) ? mean
                                    : (M1 + M2 + M3) * (1.0f / 3.0f)) * xm;  // k=3
    const float o2 = mean * xm;  // k=-1 (mean_pool; ==0 when xm==0 in ref)

    float* op = out + ((size_t)(b * Xn + x) * 3) * Cn + tid;
    op[0 * Cn] = o0;
    op[1 * Cn] = o1;
    op[2 * Cn] = o2;
  }
}

extern "C" void kernel_launch(void* const* d_in, const int* in_sizes, int n_in,
                              void* d_out, int out_size, void* d_ws, size_t ws_size,
                              hipStream_t stream) {
  const float* inp = (const float*)d_in[0];  // inp_mm (B,X,Y,C) fp32
  const float* xm  = (const float*)d_in[1];  // x_mask (B,X) fp32
  const float* ym  = (const float*)d_in[2];  // y_mask (B,Y) fp32
  float* out = (float*)d_out;                // (B,X,3,C) fp32

  dim3 grid(Bn * Xn);
  dim3 block(256);
  topk_pool_kernel<<<grid, block, 0, stream>>>(inp, xm, ym, out);
}